// HeteroSAGE_70231305224958
// MI455X (gfx1250) — compile-verified
//
#include <hip/hip_runtime.h>

// ---------------------------------------------------------------------------
// HeteroSAGE on MI455X (gfx1250, wave32).
//  * Dense transforms use V_WMMA_F32_16X16X4_F32 (f32-exact matrix pipe).
//  * Aggregation: transform-then-scatter (32 floats/edge) with f32 atomics,
//    L2-resident working set (192MB L2 >> 80MB total features).
// ---------------------------------------------------------------------------

typedef __attribute__((ext_vector_type(2))) float v2f;
typedef __attribute__((ext_vector_type(8))) float v8f;

#define NPAD(n) (((size_t)(n) + 63) & ~(size_t)63)

// ---------------- WMMA GEMM:  Y[w] = X (nrows x K) @ W[w] (K x 32) ----------
// One wave -> 16 output rows, 2 column tiles (32 cols), NW weight matrices.
// A fragment (16x4 f32): lanes 0-15 hold rows M=0..15 cols {k,k+1};
//                        lanes 16-31 hold rows M=0..15 cols {k+2,k+3}.
// B fragment (4x16 f32): lane&15 = N; VGPR0 = K rows {k | k+2}, VGPR1 = +1.
// C/D (16x16 f32): VGPR v -> row v (lanes 0-15) / row v+8 (lanes 16-31).
template <int K, int NW>
__global__ void hsage_wmma_gemm(const float* __restrict__ X, int nrows,
                                const float* __restrict__ W0,
                                const float* __restrict__ W1,
                                const float* __restrict__ W2,
                                float* __restrict__ Y0,
                                float* __restrict__ Y1,
                                float* __restrict__ Y2) {
  const int lane = threadIdx.x & 31;
  const int wave = (int)((blockIdx.x * blockDim.x + threadIdx.x) >> 5);
  const int row0 = wave << 4;
  if (row0 >= nrows) return;  // whole wave exits together -> EXEC all-1 for WMMA

  const float* Ws[3] = {W0, W1, W2};
  float*       Ys[3] = {Y0, Y1, Y2};

  const int m     = lane & 15;
  const int khalf = (lane >> 4) << 1;  // 0 or 2

  const v8f zero = {0.f, 0.f, 0.f, 0.f, 0.f, 0.f, 0.f, 0.f};
  v8f acc[NW][2];
#pragma unroll
  for (int w = 0; w < NW; ++w) {
    acc[w][0] = zero;
    acc[w][1] = zero;
  }

  const float* arow = X + (size_t)(row0 + m) * K;
#pragma unroll
  for (int k = 0; k < K; k += 4) {
    v2f a = *(const v2f*)(arow + k + khalf);  // 8B-aligned (k+khalf even)
#pragma unroll
    for (int w = 0; w < NW; ++w) {
#pragma unroll
      for (int ct = 0; ct < 2; ++ct) {
        const float* wp = Ws[w] + (size_t)(k + khalf) * 32 + ct * 16 + m;
        v2f b;
        b.x = wp[0];
        b.y = wp[32];
        acc[w][ct] = __builtin_amdgcn_wmma_f32_16x16x4_f32(
            false, a, false, b, (short)0, acc[w][ct], false, false);
      }
    }
  }

  const int rtop = row0 + ((lane >> 4) << 3);
#pragma unroll
  for (int w = 0; w < NW; ++w)
#pragma unroll
    for (int ct = 0; ct < 2; ++ct)
#pragma unroll
      for (int v = 0; v < 8; ++v)
        Ys[w][(size_t)(rtop + v) * 32 + ct * 16 + m] = acc[w][ct][v];
}

// ---------------- small weight/bias fusion ---------------------------------
__global__ void hsage_prep_weights(const float* __restrict__ Wr1pp,
                                   const float* __restrict__ Wr1ap,
                                   const float* __restrict__ Wr2pp,
                                   const float* __restrict__ Wr2ap,
                                   const float* __restrict__ b1pp,
                                   const float* __restrict__ b1ap,
                                   const float* __restrict__ b2pp,
                                   const float* __restrict__ b2ap,
                                   float* __restrict__ Wr1p,
                                   float* __restrict__ Wr2p,
                                   float* __restrict__ b1p,
                                   float* __restrict__ b2p) {
  int t = blockIdx.x * blockDim.x + threadIdx.x;
  if (t < 64 * 32) Wr1p[t] = Wr1pp[t] + Wr1ap[t];
  if (t < 32 * 32) Wr2p[t] = Wr2pp[t] + Wr2ap[t];
  if (t < 32) {
    b1p[t] = b1pp[t] + b1ap[t];
    b2p[t] = b2pp[t] + b2ap[t];
  }
}

// ---------------- degree count ---------------------------------------------
__global__ void hsage_degree(const int* __restrict__ dst, float* __restrict__ deg,
                             int nE) {
  int e = blockIdx.x * blockDim.x + threadIdx.x;
  if (e < nE) atomicAdd(&deg[dst[e]], 1.0f);
}

// ---------------- edge scatter: one wave per edge, 32 cols -----------------
__global__ void hsage_scatter32(const float* __restrict__ H,
                                const int* __restrict__ src,
                                const int* __restrict__ dst,
                                float* __restrict__ AGG, int nE) {
  int t = blockIdx.x * blockDim.x + threadIdx.x;
  int e = t >> 5;
  int c = t & 31;
  if (e >= nE) return;
  int s = src[e];
  int d = dst[e];
  atomicAdd(&AGG[(size_t)d * 32 + c], H[(size_t)s * 32 + c]);
}

// ---------------- layer-1 combines -----------------------------------------
__global__ void hsage_combine_p1(const float* __restrict__ aggpp,
                                 const float* __restrict__ aggap,
                                 const float* __restrict__ degpp,
                                 const float* __restrict__ degap,
                                 const float* __restrict__ rootp,
                                 const float* __restrict__ b1p,
                                 float* __restrict__ p1, int n) {
  int t = blockIdx.x * blockDim.x + threadIdx.x;
  if (t >= n * 32) return;
  int r = t >> 5;
  int c = t & 31;
  float v = aggpp[t] / fmaxf(degpp[r], 1.f) + aggap[t] / fmaxf(degap[r], 1.f) +
            rootp[t] + b1p[c];
  p1[t] = fmaxf(v, 0.f);
}

__global__ void hsage_combine_a1(const float* __restrict__ aggpa,
                                 const float* __restrict__ degpa,
                                 const float* __restrict__ roota,
                                 const float* __restrict__ bpa,
                                 float* __restrict__ a1, int n) {
  int t = blockIdx.x * blockDim.x + threadIdx.x;
  if (t >= n * 32) return;
  int r = t >> 5;
  int c = t & 31;
  float v = aggpa[t] / fmaxf(degpa[r], 1.f) + roota[t] + bpa[c];
  a1[t] = fmaxf(v, 0.f);
}

// ---------------- layer-2 combine + final Linear(32,1) ---------------------
__global__ void hsage_combine_p2_out(const float* __restrict__ agg2pp,
                                     const float* __restrict__ agg2ap,
                                     const float* __restrict__ degpp,
                                     const float* __restrict__ degap,
                                     const float* __restrict__ root2,
                                     const float* __restrict__ b2p,
                                     const float* __restrict__ Wlin,
                                     const float* __restrict__ blin,
                                     float* __restrict__ out, int n) {
  int t = blockIdx.x * blockDim.x + threadIdx.x;
  int r = t >> 5;
  int c = t & 31;
  if (r >= n) return;
  float v = agg2pp[t] / fmaxf(degpp[r], 1.f) + agg2ap[t] / fmaxf(degap[r], 1.f) +
            root2[t] + b2p[c];
  float s = v * Wlin[c];
#pragma unroll
  for (int off = 16; off > 0; off >>= 1) s += __shfl_xor(s, off, 32);
  if (c == 0) out[r] = s + blin[0];
}

// ---------------------------------------------------------------------------
extern "C" void kernel_launch(void* const* d_in, const int* in_sizes, int n_in,
                              void* d_out, int out_size, void* d_ws,
                              size_t ws_size, hipStream_t stream) {
  const float* x_paper  = (const float*)d_in[0];
  const float* x_author = (const float*)d_in[1];
  const int* src_pp = (const int*)d_in[2];
  const int* dst_pp = (const int*)d_in[3];
  const int* src_ap = (const int*)d_in[4];
  const int* dst_ap = (const int*)d_in[5];
  const int* src_pa = (const int*)d_in[6];
  const int* dst_pa = (const int*)d_in[7];
  const float* Wl1_pp = (const float*)d_in[8];
  const float* bl1_pp = (const float*)d_in[9];
  const float* Wr1_pp = (const float*)d_in[10];
  const float* Wl1_ap = (const float*)d_in[11];
  const float* bl1_ap = (const float*)d_in[12];
  const float* Wr1_ap = (const float*)d_in[13];
  const float* Wl1_pa = (const float*)d_in[14];
  const float* bl1_pa = (const float*)d_in[15];
  const float* Wr1_pa = (const float*)d_in[16];
  const float* Wl2_pp = (const float*)d_in[17];
  const float* bl2_pp = (const float*)d_in[18];
  const float* Wr2_pp = (const float*)d_in[19];
  const float* Wl2_ap = (const float*)d_in[20];
  const float* bl2_ap = (const float*)d_in[21];
  const float* Wr2_ap = (const float*)d_in[22];
  const float* W_lin  = (const float*)d_in[23];
  const float* b_lin  = (const float*)d_in[24];
  float* out = (float*)d_out;

  const int NP = in_sizes[0] / 64;   // 200000
  const int NA = in_sizes[1] / 64;   // 100000
  const int Epp = in_sizes[2];
  const int Eap = in_sizes[4];
  const int Epa = in_sizes[6];

  // -------- workspace arena (floats), layer-2 aliased onto layer-1 ---------
  float* ws = (float*)d_ws;
  size_t o = 0;
  auto alloc = [&](size_t n) {
    float* p = ws + o;
    o += NPAD(n);
    return p;
  };
  float* Wr1_p  = alloc(64 * 32);
  float* Wr2_p  = alloc(32 * 32);
  float* b1_p   = alloc(32);
  float* b2_p   = alloc(32);
  float* deg_pp = alloc(NP);
  float* deg_ap = alloc(NP);
  float* deg_pa = alloc(NA);
  float* deg_end = ws + o;
  float* hl_pp  = alloc((size_t)NP * 32);  // layer2 reuse: hl2_pp
  float* hl_pa  = alloc((size_t)NP * 32);  // layer2 reuse: root2
  float* root_p = alloc((size_t)NP * 32);
  float* hl_ap  = alloc((size_t)NA * 32);  // layer2 reuse: hl2_ap
  float* root_a = alloc((size_t)NA * 32);
  float* aggp_pp = alloc((size_t)NP * 32);  // layer2 reuse: agg2_pp
  float* aggp_ap = alloc((size_t)NP * 32);  // layer2 reuse: agg2_ap
  float* agga_pa = alloc((size_t)NA * 32);
  float* agg_end = ws + o;
  float* p1 = alloc((size_t)NP * 32);
  float* a1 = alloc((size_t)NA * 32);
  (void)ws_size;

  float* hl2_pp  = hl_pp;
  float* root2   = hl_pa;
  float* hl2_ap  = hl_ap;
  float* agg2_pp = aggp_pp;
  float* agg2_ap = aggp_ap;

  // -------- zero degree + layer-1 accumulators -----------------------------
  hipMemsetAsync(deg_pp, 0, (size_t)(deg_end - deg_pp) * sizeof(float), stream);
  hipMemsetAsync(aggp_pp, 0, (size_t)(agg_end - aggp_pp) * sizeof(float), stream);

  // -------- fused root weights / biases ------------------------------------
  hsage_prep_weights<<<8, 256, 0, stream>>>(Wr1_pp, Wr1_ap, Wr2_pp, Wr2_ap,
                                            bl1_pp, bl1_ap, bl2_pp, bl2_ap,
                                            Wr1_p, Wr2_p, b1_p, b2_p);

  // -------- layer-1 dense transforms (WMMA) --------------------------------
  {
    int thr = (NP / 16) * 32;
    hsage_wmma_gemm<64, 3><<<(thr + 255) / 256, 256, 0, stream>>>(
        x_paper, NP, Wl1_pp, Wl1_pa, Wr1_p, hl_pp, hl_pa, root_p);
  }
  {
    int thr = (NA / 16) * 32;
    hsage_wmma_gemm<64, 2><<<(thr + 255) / 256, 256, 0, stream>>>(
        x_author, NA, Wl1_ap, Wr1_pa, nullptr, hl_ap, root_a, nullptr);
  }

  // -------- degrees ---------------------------------------------------------
  hsage_degree<<<(Epp + 255) / 256, 256, 0, stream>>>(dst_pp, deg_pp, Epp);
  hsage_degree<<<(Eap + 255) / 256, 256, 0, stream>>>(dst_ap, deg_ap, Eap);
  hsage_degree<<<(Epa + 255) / 256, 256, 0, stream>>>(dst_pa, deg_pa, Epa);

  // -------- layer-1 scatter (transform-then-aggregate) ---------------------
  hsage_scatter32<<<(int)(((size_t)Epp * 32 + 255) / 256), 256, 0, stream>>>(
      hl_pp, src_pp, dst_pp, aggp_pp, Epp);
  hsage_scatter32<<<(int)(((size_t)Eap * 32 + 255) / 256), 256, 0, stream>>>(
      hl_ap, src_ap, dst_ap, aggp_ap, Eap);
  hsage_scatter32<<<(int)(((size_t)Epa * 32 + 255) / 256), 256, 0, stream>>>(
      hl_pa, src_pa, dst_pa, agga_pa, Epa);

  // -------- layer-1 combine + ReLU -----------------------------------------
  hsage_combine_p1<<<(int)(((size_t)NP * 32 + 255) / 256), 256, 0, stream>>>(
      aggp_pp, aggp_ap, deg_pp, deg_ap, root_p, b1_p, p1, NP);
  hsage_combine_a1<<<(int)(((size_t)NA * 32 + 255) / 256), 256, 0, stream>>>(
      agga_pa, deg_pa, root_a, bl1_pa, a1, NA);

  // -------- zero layer-2 accumulators (aliased region) ---------------------
  hipMemsetAsync(agg2_pp, 0, (size_t)(agga_pa - agg2_pp) * sizeof(float), stream);

  // -------- layer-2 dense transforms (WMMA, K=32) --------------------------
  {
    int thr = (NP / 16) * 32;
    hsage_wmma_gemm<32, 2><<<(thr + 255) / 256, 256, 0, stream>>>(
        p1, NP, Wl2_pp, Wr2_p, nullptr, hl2_pp, root2, nullptr);
  }
  {
    int thr = (NA / 16) * 32;
    hsage_wmma_gemm<32, 1><<<(thr + 255) / 256, 256, 0, stream>>>(
        a1, NA, Wl2_ap, nullptr, nullptr, hl2_ap, nullptr, nullptr);
  }

  // -------- layer-2 scatter -------------------------------------------------
  hsage_scatter32<<<(int)(((size_t)Epp * 32 + 255) / 256), 256, 0, stream>>>(
      hl2_pp, src_pp, dst_pp, agg2_pp, Epp);
  hsage_scatter32<<<(int)(((size_t)Eap * 32 + 255) / 256), 256, 0, stream>>>(
      hl2_ap, src_ap, dst_ap, agg2_ap, Eap);

  // -------- layer-2 combine + final Linear(32,1) ---------------------------
  hsage_combine_p2_out<<<(int)(((size_t)NP * 32 + 255) / 256), 256, 0, stream>>>(
      agg2_pp, agg2_ap, deg_pp, deg_ap, root2, b2_p, W_lin, b_lin, out, NP);

  (void)n_in;
  (void)out_size;
}